// god_67216238182486
// MI455X (gfx1250) — compile-verified
//
#include <hip/hip_runtime.h>
#include <math.h>

// Problem constants (fixed by the reference setup).
#define F        192        // FEAT_DIM = 3*64
#define NNODES   50000
#define NEDGES   800000
#define NCLS2    100        // 2*N_CLASSES
#define NSTEPS   6          // step_max
#define LDSW     196        // padded LDS row stride (196%64==4 -> conflict-free column reads)

typedef float v2f __attribute__((ext_vector_type(2)));
typedef float v8f __attribute__((ext_vector_type(8)));

// Full-precision fp32 WMMA: D(16x16) = A(16x4) * B(4x16) + C
__device__ __forceinline__ v8f wmma4(v2f a, v2f b, v8f c) {
  return __builtin_amdgcn_wmma_f32_16x16x4_f32(false, a, false, b, (short)0, c, false, false);
}

__device__ __forceinline__ float sigm(float x) { return 1.0f / (1.0f + expf(-x)); }

// ---------------------------------------------------------------------------
// 1) Embedding gather: h[n, p*64+e] = table[prop_ids[n,p], e]
// ---------------------------------------------------------------------------
__global__ __launch_bounds__(256) void embed_kernel(const int* __restrict__ prop_ids,
                                                    const float* __restrict__ table,
                                                    float* __restrict__ h) {
  long long idx = (long long)blockIdx.x * 256 + threadIdx.x;
  if (idx >= (long long)NNODES * F) return;
  int n = (int)(idx / F), f = (int)(idx % F);
  int p = f >> 6, e = f & 63;
  h[idx] = table[(long long)prop_ids[n * 3 + p] * 64 + e];
}

__global__ __launch_bounds__(256) void zero_kernel(float* __restrict__ p, long long n) {
  long long i = (long long)blockIdx.x * 256 + threadIdx.x;
  if (i < n) p[i] = 0.0f;
}

// ---------------------------------------------------------------------------
// 2) m = h @ W_edge^T   (M=50000, N=192, K=192), block tile 64x192, 8 waves.
//    Wave w: row subtile (w&3), column half (w>>2) covering 6 16-col subtiles.
//    K loop is software-pipelined: next-step A (LDS) + B (global, L2-hot)
//    fragments are prefetched into registers under the current 6 WMMAs.
// ---------------------------------------------------------------------------
__global__ __launch_bounds__(256) void gemm_edge_kernel(const float* __restrict__ h,
                                                        const float* __restrict__ We,
                                                        float* __restrict__ m) {
  __shared__ float sA[64 * LDSW];
  const int tid = threadIdx.x;
  const int r0 = blockIdx.x * 64;

  // Stage A tile (64 x 192) coalesced, float4 granules.
  for (int i = tid; i < 64 * 48; i += 256) {
    int r = i / 48, c4 = (i % 48) * 4;
    int row = r0 + r;
    float4 v = make_float4(0.f, 0.f, 0.f, 0.f);
    if (row < NNODES) v = *(const float4*)(h + (long long)row * F + c4);
    *(float4*)(sA + r * LDSW + c4) = v;
  }
  __syncthreads();

  const int lane = tid & 31, w = tid >> 5;
  const int lo = lane & 15, hi = lane >> 4;
  const int rsub = w & 3, ch = w >> 2;

  v8f acc[6] = {};
  const float* sArow = sA + (rsub * 16 + lo) * LDSW + 2 * hi;
  const float* Bp[6];
#pragma unroll
  for (int j = 0; j < 6; j++)
    Bp[j] = We + (long long)((ch * 6 + j) * 16 + lo) * F + 2 * hi;

  // Prologue: fragments for kb = 0.
  v2f an = *(const v2f*)(sArow);
  v2f bn[6];
#pragma unroll
  for (int j = 0; j < 6; j++) bn[j] = *(const v2f*)(Bp[j]);

#pragma unroll 4
  for (int kb = 0; kb < F - 4; kb += 4) {
    v2f a0 = an;
    v2f b0[6];
#pragma unroll
    for (int j = 0; j < 6; j++) b0[j] = bn[j];
    // Prefetch kb+4 while the 6 WMMAs below execute.
    an = *(const v2f*)(sArow + kb + 4);
#pragma unroll
    for (int j = 0; j < 6; j++) bn[j] = *(const v2f*)(Bp[j] + kb + 4);
#pragma unroll
    for (int j = 0; j < 6; j++) acc[j] = wmma4(a0, b0[j], acc[j]);
  }
#pragma unroll
  for (int j = 0; j < 6; j++) acc[j] = wmma4(an, bn[j], acc[j]);   // epilogue

  __syncthreads();                               // done reading sA; reuse for output
#pragma unroll
  for (int j = 0; j < 6; j++) {
    int col = (ch * 6 + j) * 16 + lo;
#pragma unroll
    for (int v = 0; v < 8; v++) {
      int rl = rsub * 16 + v + 8 * hi;
      sA[rl * LDSW + col] = acc[j][v];
    }
  }
  __syncthreads();
  for (int i = tid; i < 64 * F; i += 256) {
    int r = i / F, c = i % F;
    int row = r0 + r;
    if (row < NNODES) m[(long long)row * F + c] = sA[r * LDSW + c];
  }
}

// ---------------------------------------------------------------------------
// 3) a[dst[e], :] += m[src[e], :]  -- float4 granules: one global_load_b128 +
//    four global_atomic_add_f32 per thread; atomics resolve in the 192MB L2.
// ---------------------------------------------------------------------------
__global__ __launch_bounds__(256) void scatter_kernel(const float* __restrict__ m,
                                                      const int* __restrict__ src,
                                                      const int* __restrict__ dst,
                                                      float* __restrict__ a) {
  long long i = ((long long)blockIdx.x * 256 + threadIdx.x) * 4;
  if (i >= (long long)NEDGES * F) return;
  int e = (int)(i / F);
  int f = (int)(i % F);
  int s = src[e], d = dst[e];
  float4 v = *(const float4*)(m + (long long)s * F + f);
  float* ap = a + (long long)d * F + f;
  atomicAdd(ap + 0, v.x);
  atomicAdd(ap + 1, v.y);
  atomicAdd(ap + 2, v.z);
  atomicAdd(ap + 3, v.w);
}

// ---------------------------------------------------------------------------
// 4) Fused GRU cell: gi = a@Wih^T + bih, gh = h@Whh^T + bhh, gates, h_new.
//    Block: 64 rows x 32 cols (of F), grid.y = F/32 = 6.
//    Wave: 16x16 col tile -> 6 accumulators (r,z,n for gi and gh).
//    K loop software-pipelined (2 LDS A frags + 6 global B frags prefetched).
// ---------------------------------------------------------------------------
__global__ __launch_bounds__(256) void gru_kernel(const float* __restrict__ a,
                                                  const float* __restrict__ hprev,
                                                  const float* __restrict__ Wih,
                                                  const float* __restrict__ Whh,
                                                  const float* __restrict__ bih,
                                                  const float* __restrict__ bhh,
                                                  float* __restrict__ hnext) {
  __shared__ float sA[64 * LDSW];
  __shared__ float sH[64 * LDSW];
  const int tid = threadIdx.x;
  const int r0 = blockIdx.x * 64;
  const int CB = blockIdx.y * 32;

  for (int i = tid; i < 64 * 48; i += 256) {
    int r = i / 48, c4 = (i % 48) * 4;
    int row = r0 + r;
    float4 va = make_float4(0.f, 0.f, 0.f, 0.f), vh = va;
    if (row < NNODES) {
      va = *(const float4*)(a + (long long)row * F + c4);
      vh = *(const float4*)(hprev + (long long)row * F + c4);
    }
    *(float4*)(sA + r * LDSW + c4) = va;
    *(float4*)(sH + r * LDSW + c4) = vh;
  }
  __syncthreads();

  const int lane = tid & 31, w = tid >> 5;
  const int lo = lane & 15, hi = lane >> 4;
  const int rsub = w & 3, ch = w >> 2;
  const int cb = CB + ch * 16;

  v8f acc[6] = {};   // 0:gi_r 1:gi_z 2:gi_n 3:gh_r 4:gh_z 5:gh_n
  const float* sArow = sA + (rsub * 16 + lo) * LDSW + 2 * hi;
  const float* sHrow = sH + (rsub * 16 + lo) * LDSW + 2 * hi;
  const float* Bp[6];
  Bp[0] = Wih + (long long)(cb + lo) * F + 2 * hi;
  Bp[1] = Wih + (long long)(cb + 192 + lo) * F + 2 * hi;
  Bp[2] = Wih + (long long)(cb + 384 + lo) * F + 2 * hi;
  Bp[3] = Whh + (long long)(cb + lo) * F + 2 * hi;
  Bp[4] = Whh + (long long)(cb + 192 + lo) * F + 2 * hi;
  Bp[5] = Whh + (long long)(cb + 384 + lo) * F + 2 * hi;

  v2f aan = *(const v2f*)(sArow);
  v2f ahn = *(const v2f*)(sHrow);
  v2f bn[6];
#pragma unroll
  for (int j = 0; j < 6; j++) bn[j] = *(const v2f*)(Bp[j]);

#pragma unroll 4
  for (int kb = 0; kb < F - 4; kb += 4) {
    v2f aa = aan, ah = ahn;
    v2f b0[6];
#pragma unroll
    for (int j = 0; j < 6; j++) b0[j] = bn[j];
    aan = *(const v2f*)(sArow + kb + 4);
    ahn = *(const v2f*)(sHrow + kb + 4);
#pragma unroll
    for (int j = 0; j < 6; j++) bn[j] = *(const v2f*)(Bp[j] + kb + 4);
    acc[0] = wmma4(aa, b0[0], acc[0]);
    acc[1] = wmma4(aa, b0[1], acc[1]);
    acc[2] = wmma4(aa, b0[2], acc[2]);
    acc[3] = wmma4(ah, b0[3], acc[3]);
    acc[4] = wmma4(ah, b0[4], acc[4]);
    acc[5] = wmma4(ah, b0[5], acc[5]);
  }
  acc[0] = wmma4(aan, bn[0], acc[0]);
  acc[1] = wmma4(aan, bn[1], acc[1]);
  acc[2] = wmma4(aan, bn[2], acc[2]);
  acc[3] = wmma4(ahn, bn[3], acc[3]);
  acc[4] = wmma4(ahn, bn[4], acc[4]);
  acc[5] = wmma4(ahn, bn[5], acc[5]);

  const float bir = bih[cb + lo], biz = bih[cb + 192 + lo], bin = bih[cb + 384 + lo];
  const float bhr = bhh[cb + lo], bhz = bhh[cb + 192 + lo], bhn = bhh[cb + 384 + lo];

  float hv[8];
#pragma unroll
  for (int v = 0; v < 8; v++) {
    int rl = rsub * 16 + v + 8 * hi;
    float r_ = sigm(acc[0][v] + bir + acc[3][v] + bhr);
    float z_ = sigm(acc[1][v] + biz + acc[4][v] + bhz);
    float n_ = tanhf(acc[2][v] + bin + r_ * (acc[5][v] + bhn));
    float hold = sH[rl * LDSW + cb + lo];        // h_old from staged LDS tile
    hv[v] = (1.0f - z_) * n_ + z_ * hold;
  }

  __syncthreads();                               // all waves done reading sA
#pragma unroll
  for (int v = 0; v < 8; v++) {
    int rl = rsub * 16 + v + 8 * hi;
    sA[rl * 36 + ch * 16 + lo] = hv[v];          // 64x32 out tile, stride 36
  }
  __syncthreads();
  for (int i = tid; i < 64 * 32; i += 256) {
    int r = i / 32, c = i % 32;
    int row = r0 + r;
    if (row < NNODES) hnext[(long long)row * F + CB + c] = sA[r * 36 + c];
  }
}

// ---------------------------------------------------------------------------
// 5) Per-step classifier: out[(n*5+s)*100+c] = sigmoid(h·conv_w[c] + conv_b[c])
//    N padded to 128 (8 col subtiles); block = 64 rows, 8 waves, 4 tiles each.
// ---------------------------------------------------------------------------
__global__ __launch_bounds__(256) void conv_kernel(const float* __restrict__ hs,
                                                   const float* __restrict__ cw,
                                                   const float* __restrict__ cbias,
                                                   float* __restrict__ out, int sidx) {
  __shared__ float sF[64 * LDSW];
  const int tid = threadIdx.x;
  const int r0 = blockIdx.x * 64;

  for (int i = tid; i < 64 * 48; i += 256) {
    int r = i / 48, c4 = (i % 48) * 4;
    int row = r0 + r;
    float4 v = make_float4(0.f, 0.f, 0.f, 0.f);
    if (row < NNODES) v = *(const float4*)(hs + (long long)row * F + c4);
    *(float4*)(sF + r * LDSW + c4) = v;
  }
  __syncthreads();

  const int lane = tid & 31, w = tid >> 5;
  const int lo = lane & 15, hi = lane >> 4;
  const int rsub = w & 3, ch = w >> 2;

  v8f acc[4] = {};
  const float* sFrow = sF + (rsub * 16 + lo) * LDSW + 2 * hi;
  const float* Bp[4];
  bool bval[4];
#pragma unroll
  for (int j = 0; j < 4; j++) {
    int col = (ch * 4 + j) * 16 + lo;
    bval[j] = (col < NCLS2);
    Bp[j] = cw + (long long)(bval[j] ? col : 0) * F + 2 * hi;
  }

  v2f an = *(const v2f*)(sFrow);
  v2f bn[4];
#pragma unroll
  for (int j = 0; j < 4; j++) { v2f t = *(const v2f*)(Bp[j]); bn[j] = bval[j] ? t : (v2f){0.f, 0.f}; }

#pragma unroll 4
  for (int kb = 0; kb < F - 4; kb += 4) {
    v2f a0 = an;
    v2f b0[4];
#pragma unroll
    for (int j = 0; j < 4; j++) b0[j] = bn[j];
    an = *(const v2f*)(sFrow + kb + 4);
#pragma unroll
    for (int j = 0; j < 4; j++) {
      v2f t = *(const v2f*)(Bp[j] + kb + 4);
      bn[j] = bval[j] ? t : (v2f){0.f, 0.f};
    }
#pragma unroll
    for (int j = 0; j < 4; j++) acc[j] = wmma4(a0, b0[j], acc[j]);
  }
#pragma unroll
  for (int j = 0; j < 4; j++) acc[j] = wmma4(an, bn[j], acc[j]);

  __syncthreads();                               // reuse sF as 64x100 out tile (stride 104)
#pragma unroll
  for (int j = 0; j < 4; j++) {
    int col = (ch * 4 + j) * 16 + lo;
    if (col < NCLS2) {
      float bc = cbias[col];
#pragma unroll
      for (int v = 0; v < 8; v++) {
        int rl = rsub * 16 + v + 8 * hi;
        sF[rl * 104 + col] = sigm(acc[j][v] + bc);
      }
    }
  }
  __syncthreads();
  for (int i = tid; i < 64 * NCLS2; i += 256) {
    int r = i / NCLS2, c = i % NCLS2;
    int row = r0 + r;
    if (row < NNODES)
      out[((long long)row * 5 + sidx) * NCLS2 + c] = sF[r * 104 + c];
  }
}

// ---------------------------------------------------------------------------
// Driver: 6 GRU steps with ping/pong h; conv fused after steps 2..6 so no
// snapshot storage is needed.  ws layout: h0 | h1 | m | a  (4 x 38.4 MB).
// ---------------------------------------------------------------------------
extern "C" void kernel_launch(void* const* d_in, const int* in_sizes, int n_in,
                              void* d_out, int out_size, void* d_ws, size_t ws_size,
                              hipStream_t stream) {
  const int*   prop_ids = (const int*)d_in[0];
  const int*   src      = (const int*)d_in[1];
  const int*   dst      = (const int*)d_in[2];
  const float* table    = (const float*)d_in[3];
  const float* We       = (const float*)d_in[4];
  const float* Wih      = (const float*)d_in[5];
  const float* Whh      = (const float*)d_in[6];
  const float* bih      = (const float*)d_in[7];
  const float* bhh      = (const float*)d_in[8];
  const float* cw       = (const float*)d_in[9];
  const float* cbias    = (const float*)d_in[10];
  float* out = (float*)d_out;

  const long long NF = (long long)NNODES * F;
  float* h0 = (float*)d_ws;
  float* h1 = h0 + NF;
  float* mb = h1 + NF;
  float* ab = mb + NF;

  const int rowBlocks = (NNODES + 63) / 64;                    // 782
  const int nfBlocks  = (int)((NF + 255) / 256);               // 37500
  const int edBlocks  = (int)(((long long)NEDGES * F / 4 + 255) / 256); // 150000

  embed_kernel<<<nfBlocks, 256, 0, stream>>>(prop_ids, table, h0);

  float* hcur = h0;
  float* hnxt = h1;
  for (int s = 0; s < NSTEPS; s++) {
    gemm_edge_kernel<<<rowBlocks, 256, 0, stream>>>(hcur, We, mb);
    zero_kernel<<<nfBlocks, 256, 0, stream>>>(ab, NF);
    scatter_kernel<<<edBlocks, 256, 0, stream>>>(mb, src, dst, ab);
    gru_kernel<<<dim3(rowBlocks, 6), 256, 0, stream>>>(ab, hcur, Wih, Whh, bih, bhh, hnxt);
    if (s >= 1)  // feat_steps[1..5] = h after steps 2..6
      conv_kernel<<<rowBlocks, 256, 0, stream>>>(hnxt, cw, cbias, out, s - 1);
    float* t = hcur; hcur = hnxt; hnxt = t;
  }
}